// GLCM_45432164057297
// MI455X (gfx1250) — compile-verified
//
#include <hip/hip_runtime.h>
#include <stdint.h>

// ---------------------------------------------------------------------------
// GLCM on sign-binarized image, 12 offsets, symmetric+normed.
// Memory-bound: 134MB in / 24KB out -> ~5.8us floor at 23.3 TB/s.
// CDNA5 path: TDM (tensor_load_to_lds) async staging into LDS, wave32 ballot
// bit-packing, popcount counting. WMMA intentionally unused (would inflate
// operand traffic 8x for trivial FLOPs).
// ---------------------------------------------------------------------------

typedef __attribute__((ext_vector_type(4))) unsigned int u32x4;
typedef __attribute__((ext_vector_type(8))) unsigned int u32x8;

#define HH 512
#define WW 512
#define NB 128
#define NOFF 12
#define SLICES 4
#define ROWS_PER_SLICE (HH / SLICES) /* 128 */
#define HALO 3
#define MROWS (ROWS_PER_SLICE + HALO) /* 131 */
#define CH 12                          /* rows staged per TDM op (24KB) */
#define WORDS (WW / 32)                /* 16 mask words per row */

__device__ __constant__ int OFF_R[NOFF] = {0, 1, 1, 1, 0, 2, 2, 2, 0, 3, 3, 3};
__device__ __constant__ int OFF_C[NOFF] = {1, 1, 0, -1, 2, 2, 0, -2, 3, 3, 0, -3};

__device__ __forceinline__ unsigned ballot32(bool p) {
#if defined(__has_builtin)
#if __has_builtin(__builtin_amdgcn_ballot_w32)
  return __builtin_amdgcn_ballot_w32(p);
#else
  return (unsigned)__ballot(p);
#endif
#else
  return (unsigned)__ballot(p);
#endif
}

// Issue a TDM load of an h x 512 float tile (row-major, stride 512) from
// global memory to LDS. D# built per CDNA5 ISA:
//  group0: count=1, lds_addr, global_addr[56:0], type=2
//  group1: data_size=4B, tensor_dim0=512, tensor_dim1=512,
//          tile_dim0=512, tile_dim1=h, tensor_dim0_stride=512
//  groups 2/3: zero (2-D tensor; tile_dim2=0 so higher dims unused)
__device__ __forceinline__ void tdm_load_rows(const float* gsrc,
                                              unsigned lds_byte_addr, int h) {
  unsigned long long ga = (unsigned long long)(uintptr_t)gsrc;
  u32x4 g0;
  g0[0] = 1u;                                   // count=1, is_restore=0
  g0[1] = lds_byte_addr;                        // lds_addr (bytes)
  g0[2] = (unsigned)ga;                         // global_addr[31:0]
  g0[3] = ((unsigned)(ga >> 32) & 0x01FFFFFFu)  // global_addr[56:32]
          | (2u << 30);                         // type=2 ("image")
  u32x8 g1;
  g1[0] = (2u << 16);                           // wg_mask=0, data_size=2 (4B)
  g1[1] = ((unsigned)WW & 0xFFFFu) << 16;       // tensor_dim0[15:0] in [31:16]
  g1[2] = (((unsigned)WW >> 16) & 0xFFFFu)      // tensor_dim0[31:16]
          | (((unsigned)HH & 0xFFFFu) << 16);   // tensor_dim1[15:0]
  g1[3] = (((unsigned)HH >> 16) & 0xFFFFu)      // tensor_dim1[31:16]
          | (((unsigned)WW & 0xFFFFu) << 16);   // tile_dim0 = 512
  g1[4] = (unsigned)h & 0xFFFFu;                // tile_dim1 = h, tile_dim2 = 0
  g1[5] = (unsigned)WW;                         // tensor_dim0_stride low32
  g1[6] = 0u;                                   // stride hi / dim1_stride lo
  g1[7] = 0u;                                   // dim1_stride hi
  u32x4 gz;
  gz[0] = 0u; gz[1] = 0u; gz[2] = 0u; gz[3] = 0u;
  asm volatile("tensor_load_to_lds %0, %1, %2, %3"
               :
               : "s"(g0), "s"(g1), "s"(gz), "s"(gz)
               : "memory");
}

extern "C" __global__ __launch_bounds__(1024) void glcm_count_kernel(
    const float* __restrict__ in, unsigned* __restrict__ ws) {
  __shared__ __align__(16) float fbuf[2][CH * WW];       // 2 x 24 KB staging
  __shared__ unsigned masks[MROWS][WORDS + 1];           // sign bit masks
  __shared__ unsigned acc[NOFF * 2];                     // per-offset mm,cross

  const int tid = threadIdx.x;
  const int wave = tid >> 5;
  const int lane = tid & 31;
  const int slice = blockIdx.x;
  const int batch = blockIdx.y;
  const int r0 = slice * ROWS_PER_SLICE;
  const int nrows = min(MROWS, HH - r0);                 // rows to bit-pack
  const int nchunks = (nrows + CH - 1) / CH;
  const float* gsrc = in + ((size_t)batch * HH + r0) * WW;

  if (tid < NOFF * 2) acc[tid] = 0u;

  // ---- Phase 1: TDM-stage rows, convert to 1-bit sign masks via ballot ----
  if (tid < 32) {
    tdm_load_rows(gsrc, (unsigned)(uintptr_t)&fbuf[0][0], min(CH, nrows));
  }
  for (int ck = 0; ck < nchunks; ++ck) {
    const int h = min(CH, nrows - ck * CH);
    if (tid < 32) {
      if (ck + 1 < nchunks) {
        tdm_load_rows(gsrc + (size_t)(ck + 1) * CH * WW,
                      (unsigned)(uintptr_t)&fbuf[(ck + 1) & 1][0],
                      min(CH, nrows - (ck + 1) * CH));
        asm volatile("s_wait_tensorcnt 0x1" ::: "memory"); // oldest done
      } else {
        asm volatile("s_wait_tensorcnt 0x0" ::: "memory");
      }
    }
    __syncthreads();
    const float* fb = fbuf[ck & 1];
    const int ngroups = h * WORDS; // 32-pixel groups in this chunk
    for (int g = wave; g < ngroups; g += 32) {
      const float v = fb[g * 32 + lane];
      const unsigned m = ballot32(v > 0.0f); // bit i == sign(col 32w+i) > 0
      if (lane == 0) masks[ck * CH + (g >> 4)][g & (WORDS - 1)] = m;
    }
    __syncthreads();
  }

  // ---- Phase 2: per (offset,row) count mixed pairs and --- pairs ----------
  for (int t = tid; t < NOFF * ROWS_PER_SLICE; t += 1024) {
    const int o = t >> 7;  // /128
    const int y = t & 127; // local a-row
    const int r = OFF_R[o];
    const int c = OFF_C[o];
    if (r0 + y + r >= HH) continue;
    unsigned mm = 0, cross = 0;
    const unsigned* ma = masks[y];
    const unsigned* mb = masks[y + r];
    const int s = 32 + c; // extract 32 bits starting at bit 32+c of window
#pragma unroll
    for (int w = 0; w < WORDS; ++w) {
      const unsigned a = ma[w];
      const unsigned lo = (w > 0) ? mb[w - 1] : 0u;
      const unsigned mi = mb[w];
      const unsigned hi = (w < WORDS - 1) ? mb[w + 1] : 0u;
      const unsigned long long lo64 =
          (unsigned long long)lo | ((unsigned long long)mi << 32);
      const unsigned b =
          (unsigned)((lo64 >> s) | ((unsigned long long)hi << (64 - s)));
      unsigned valid = 0xFFFFFFFFu;
      if (c > 0 && w == WORDS - 1) valid = 0xFFFFFFFFu >> c;  // x <= 511-c
      if (c < 0 && w == 0) valid = 0xFFFFFFFFu << (-c);       // x >= |c|
      cross += __popc((a ^ b) & valid);
      mm += __popc((~(a | b)) & valid);
    }
    atomicAdd(&acc[o * 2 + 0], mm);
    atomicAdd(&acc[o * 2 + 1], cross);
  }
  __syncthreads();
  if (tid < NOFF * 2) atomicAdd(&ws[batch * (NOFF * 2) + tid], acc[tid]);
}

// ---- Phase 3: integer counts -> symmetric, normalized GLCM floats ---------
extern "C" __global__ void glcm_finalize_kernel(const unsigned* __restrict__ ws,
                                                float* __restrict__ out) {
  const int t = blockIdx.x * blockDim.x + threadIdx.x;
  if (t >= NB * NOFF) return;
  const int b = t / NOFF;
  const int o = t % NOFF;
  const int r = OFF_R[o];
  const int c = OFF_C[o];
  const int ac = c < 0 ? -c : c;
  const float N = (float)((HH - r) * (WW - ac));
  const float mm = (float)ws[b * (NOFF * 2) + o * 2];
  const float cross = (float)ws[b * (NOFF * 2) + o * 2 + 1];
  const float npp = N - mm - cross;
  // M00=2mm, M01+M10=2cross-4mm, M11=npp-cross+mm; temp=[2M00,S,S,2M11]
  const float t0 = 4.0f * mm;
  const float t1 = 2.0f * cross - 4.0f * mm;
  const float t3 = 2.0f * (npp - cross + mm);
  const float inv = 1.0f / (t0 + 2.0f * t1 + t3);
  float* op = out + (b * NOFF + o) * 4;
  op[0] = t0 * inv;
  op[1] = t1 * inv;
  op[2] = t1 * inv;
  op[3] = t3 * inv;
}

extern "C" void kernel_launch(void* const* d_in, const int* in_sizes, int n_in,
                              void* d_out, int out_size, void* d_ws,
                              size_t ws_size, hipStream_t stream) {
  const float* in = (const float*)d_in[0];
  float* out = (float*)d_out;
  unsigned* ws = (unsigned*)d_ws;
  (void)in_sizes; (void)n_in; (void)out_size; (void)ws_size;

  hipMemsetAsync(ws, 0, NB * NOFF * 2 * sizeof(unsigned), stream);
  dim3 grid(SLICES, NB);
  hipLaunchKernelGGL(glcm_count_kernel, grid, dim3(1024), 0, stream, in, ws);
  hipLaunchKernelGGL(glcm_finalize_kernel, dim3((NB * NOFF + 255) / 256),
                     dim3(256), 0, stream, ws, out);
}